// ChamferDistanceLoss_25074019074268
// MI455X (gfx1250) — compile-verified
//
#include <hip/hip_runtime.h>

// CDNA5 (gfx1250) Chamfer distance, WMMA-f32 path, v2.
//
// dist tile via ONE V_WMMA_F32_16X16X4_F32 per 16x16 (row,col) block, with
// all per-tile operand construction minimized:
//   A row i = (-2x_i, -2y_i, -2z_i, 1)    <- hoisted, computed once per wave
//   B col j = ( x'_j,  y'_j,  z'_j, |q_j|^2)
//   C       = 0                           <- loop-invariant (inline-0 / hoisted)
//   D = A*B + C = |q_j|^2 - 2 p_i.q_j
// min_j(|p|^2 + |q|^2 - 2p.q) = |p|^2 + min_j(|q|^2 - 2p.q), so |p_i|^2 is
// added once per row AFTER the min scan (8 shfls per wave), not per tile.
// Per column tile this leaves: 1 b96 load, 3 VALU (sq2), 2 cndmask, 1 WMMA,
// amortized min3s — WMMA-dense instead of mov-dense.
//
// Column-direction mins come from a second symmetric pass with P/Q swapped
// (avoids 16M global float atomics, keeps everything deterministic). Inputs
// (1.6 MB total) live in WGP$/L2, so both passes are compute-bound.

typedef float v2f __attribute__((ext_vector_type(2)));
typedef float v8f __attribute__((ext_vector_type(8)));

__device__ __forceinline__ float fmin_num(float a, float b) {
    return __builtin_fminf(a, b);
}

// For each (batch, 16-row tile of P): rowmin[b, i] = min_j dist(P[b,i], Q[b,j])
__global__ __launch_bounds__(128) void chamfer_rowmin_kernel(
    const float* __restrict__ P,   // (B, NP, 3)
    const float* __restrict__ Q,   // (B, NQ, 3)
    float* __restrict__ rowmin,    // (B, NP)
    int B, int NP, int NQ)
{
    const int lane        = threadIdx.x & 31;
    const int waveInBlock = threadIdx.x >> 5;
    const int wid         = blockIdx.x * 4 + waveInBlock;   // global wave id
    const int tilesPerB   = NP >> 4;
    const int b           = wid / tilesPerB;
    const int rowTile     = wid % tilesPerB;
    if (b >= B) return;  // wave-uniform exit; EXEC stays all-1 for WMMA

    const int    rowBase = rowTile << 4;
    const float* Pb = P + (size_t)b * NP * 3;
    const float* Qb = Q + (size_t)b * NQ * 3;

    // ---- A matrix (16x4 f32): lane L, VGPR v -> M = L%16, K = v + 2*(L>=16)
    // A row = (-2x, -2y, -2z, 1): the -2 scale lives here, hoisted.
    const int  r  = lane & 15;
    const bool hi = lane >= 16;
    const float px = Pb[(rowBase + r) * 3 + 0];
    const float py = Pb[(rowBase + r) * 3 + 1];
    const float pz = Pb[(rowBase + r) * 3 + 2];
    const float sq1 = px * px + py * py + pz * pz;   // added back after the min
    v2f a;
    a.x = hi ? (-2.0f * pz) : (-2.0f * px);   // K = 2 : 0
    a.y = hi ? 1.0f         : (-2.0f * py);   // K = 3 : 1

    // Loop-invariant zero C: compiler keeps these as 8 zero VGPRs (or folds
    // to the ISA's inline SRC2=0) instead of re-materializing per tile.
    const v8f czero = {};

    float rmin[8];
#pragma unroll
    for (int i = 0; i < 8; ++i) rmin[i] = 3.402823466e38f;

    for (int colBase = 0; colBase < NQ; colBase += 16) {
        // ---- B matrix (4x16 f32): lane L, VGPR v -> N = L%16, K = v + 2*(L>=16)
        // B col = (x', y', z', |q|^2)
        const float qx = Qb[(colBase + r) * 3 + 0];
        const float qy = Qb[(colBase + r) * 3 + 1];
        const float qz = Qb[(colBase + r) * 3 + 2];
        const float sq2 = qx * qx + qy * qy + qz * qz;
        v2f bm;
        bm.x = hi ? qz  : qx;   // K = 2 : 0
        bm.y = hi ? sq2 : qy;   // K = 3 : 1

        // D = A*B + 0 = |q_j|^2 - 2 p_i.q_j for the whole 16x16 tile.
        v8f d = __builtin_amdgcn_wmma_f32_16x16x4_f32(
            /*neg_a=*/false, a, /*neg_b=*/false, bm,
            /*c_mod=*/(short)0, czero, /*reuse_a=*/false, /*reuse_b=*/false);

#pragma unroll
        for (int i = 0; i < 8; ++i) rmin[i] = fmin_num(rmin[i], d[i]);
    }

    // Cross-lane min inside each 16-lane half (xor masks 1..8 stay in-half).
    // D layout: VGPR v, lane L -> row = v + 8*(L>=16), col = L%16.
#pragma unroll
    for (int m = 1; m <= 8; m <<= 1) {
#pragma unroll
        for (int i = 0; i < 8; ++i)
            rmin[i] = fmin_num(rmin[i], __shfl_xor(rmin[i], m, 32));
    }

    // Add back |p_row|^2: row (8*hi + i)'s sq1 lives in lane (8*hi + i).
    // Executed by all lanes (shfl is wave-uniform); each half adds its rows.
    const int srcBase = hi ? 8 : 0;
#pragma unroll
    for (int i = 0; i < 8; ++i)
        rmin[i] += __shfl(sq1, srcBase + i, 32);

    // Lane 0 writes rows rowBase+0..7, lane 16 writes rows rowBase+8..15.
    if (lane == 0 || lane == 16) {
        float* out = rowmin + (size_t)b * NP + rowBase + srcBase;
#pragma unroll
        for (int i = 0; i < 8; ++i) out[i] = rmin[i];
    }
}

// Sum both row-min arrays and scale by 1/B. Single workgroup, deterministic.
__global__ __launch_bounds__(1024) void chamfer_reduce_kernel(
    const float* __restrict__ mins, int n, float invB, float* __restrict__ out)
{
    __shared__ float smem[1024];
    float s = 0.0f;
    for (int i = threadIdx.x; i < n; i += 1024) s += mins[i];
    smem[threadIdx.x] = s;
    __syncthreads();
    for (int off = 512; off > 0; off >>= 1) {
        if ((int)threadIdx.x < off) smem[threadIdx.x] += smem[threadIdx.x + off];
        __syncthreads();
    }
    if (threadIdx.x == 0) out[0] = smem[0] * invB;
}

extern "C" void kernel_launch(void* const* d_in, const int* in_sizes, int n_in,
                              void* d_out, int out_size, void* d_ws, size_t ws_size,
                              hipStream_t stream) {
    const float* p1 = (const float*)d_in[0];  // (B, N, 3) f32
    const float* p2 = (const float*)d_in[1];  // (B, M, 3) f32
    const int B = 16;
    const int N = in_sizes[0] / (B * 3);
    const int M = in_sizes[1] / (B * 3);

    float* rowmin1 = (float*)d_ws;              // B*N floats
    float* rowmin2 = rowmin1 + (size_t)B * N;   // B*M floats

    // 4 waves (128 threads) per block, one 16-row tile per wave.
    const int waves1 = B * (N / 16);
    chamfer_rowmin_kernel<<<waves1 / 4, 128, 0, stream>>>(p1, p2, rowmin1, B, N, M);
    const int waves2 = B * (M / 16);
    chamfer_rowmin_kernel<<<waves2 / 4, 128, 0, stream>>>(p2, p1, rowmin2, B, M, N);

    chamfer_reduce_kernel<<<1, 1024, 0, stream>>>(rowmin1, B * (N + M),
                                                  1.0f / (float)B, (float*)d_out);
}